// ModePool2d_4492535792463
// MI455X (gfx1250) — compile-verified
//
#include <hip/hip_runtime.h>
#include <cstdint>

// Mode-pool 3x3, stride 2, pad 1 over (2,64,512,512) fp32 with integer values 0..15.
// Output (2,64,256,256). HBM-bound: stage input tiles in LDS via CDNA5 async
// global->LDS loads (ASYNCcnt), compute mode via a 16-nibble histogram in a u64.

#define B_PLANES 128   // B*C = 2*64
#define IN_H 512
#define IN_W 512
#define OUT_H 256
#define OUT_W 256

#define TILE_OW 64     // output tile width
#define TILE_OH 8      // output tile height
#define IN_TW (2 * TILE_OW + 1)   // 129 input cols per tile
#define IN_TH (2 * TILE_OH + 1)   // 17 input rows per tile
#define LDS_STRIDE 132            // padded row stride (floats); lanes hit distinct banks

__global__ __launch_bounds__(256)
void mode_pool2d_kernel(const float* __restrict__ x, float* __restrict__ out) {
    __shared__ float tile[IN_TH * LDS_STRIDE];

    const int plane = blockIdx.z;                 // 0..127  (b*C + c)
    const int ow0   = blockIdx.x * TILE_OW;       // 0,64,128,192
    const int oh0   = blockIdx.y * TILE_OH;       // 0..248
    const int ih0   = 2 * oh0 - 1;                // top input row of tile (may be -1)
    const int iw0   = 2 * ow0 - 1;                // left input col of tile (may be -1)
    const int tid   = threadIdx.x;

    const float* __restrict__ src = x + (size_t)plane * (IN_H * IN_W);

    // Group-relative LDS byte offset of the tile base (addrspace(3) ptrtoint).
    uint32_t lds_base =
        (uint32_t)(uintptr_t)(__attribute__((address_space(3))) float*)tile;

    // ---- Async copy: global -> LDS (clamped source addresses for pad region) ----
    for (int idx = tid; idx < IN_TH * IN_TW; idx += 256) {
        int r = idx / IN_TW;
        int c = idx - r * IN_TW;
        int ih = ih0 + r; if (ih < 0) ih = 0;     // only top edge can be OOB
        int iw = iw0 + c; if (iw < 0) iw = 0;     // only left edge can be OOB
        uint32_t goff = (uint32_t)((ih << 9) + iw) << 2;               // byte offset
        uint32_t loff = lds_base + (uint32_t)(r * LDS_STRIDE + c) * 4u; // LDS byte addr
        asm volatile("global_load_async_to_lds_b32 %0, %1, %2"
                     :
                     : "v"(loff), "v"(goff), "s"(src)
                     : "memory");
    }
    asm volatile("s_wait_asynccnt 0" ::: "memory");   // my wave's async ops landed
    __syncthreads();                                   // all waves' ops landed

    // ---- Compute: each thread produces 2 outputs (tile is 64x8 = 512 outputs) ----
    const int lx  = tid & 63;   // 0..63
    const int lyb = tid >> 6;   // 0..3 -> rows lyb and lyb+4

#pragma unroll
    for (int pass = 0; pass < 2; ++pass) {
        const int ly = lyb + pass * 4;            // 0..7
        uint64_t hist = 0;                        // 16 nibble counters, counts <= 9

#pragma unroll
        for (int ky = 0; ky < 3; ++ky) {
            const int r = 2 * ly + ky;
            const bool rpad = (ih0 + r) < 0;
#pragma unroll
            for (int kx = 0; kx < 3; ++kx) {
                const int c = 2 * lx + kx;
                const bool cpad = (iw0 + c) < 0;
                float v = tile[r * LDS_STRIDE + c];
                int iv = (rpad || cpad) ? 0 : (int)v;   // pad contributes value 0
                hist += 1ull << (iv << 2);
            }
        }

        // Nibble scan; strict '>' => smallest value wins ties (matches reference).
        int best = 0;
        uint32_t bc = (uint32_t)hist & 15u;
#pragma unroll
        for (int v = 1; v < 16; ++v) {
            uint32_t cnt = (uint32_t)(hist >> (v * 4)) & 15u;
            if (cnt > bc) { bc = cnt; best = v; }
        }

        out[(size_t)plane * (OUT_H * OUT_W) + (size_t)(oh0 + ly) * OUT_W + (ow0 + lx)]
            = (float)best;
    }
}

extern "C" void kernel_launch(void* const* d_in, const int* in_sizes, int n_in,
                              void* d_out, int out_size, void* d_ws, size_t ws_size,
                              hipStream_t stream) {
    (void)in_sizes; (void)n_in; (void)out_size; (void)d_ws; (void)ws_size;
    const float* x = (const float*)d_in[0];
    float* out = (float*)d_out;

    dim3 grid(OUT_W / TILE_OW, OUT_H / TILE_OH, B_PLANES);  // (4, 32, 128)
    dim3 block(256);
    hipLaunchKernelGGL(mode_pool2d_kernel, grid, block, 0, stream, x, out);
}